// Top_k_features_71992241815585
// MI455X (gfx1250) — compile-verified
//
#include <hip/hip_runtime.h>
#include <hip/hip_bf16.h>

#define N_NODES 2048
#define N_ATTRS 64
#define TOPK    8
#define FPT     4                      // features per thread
#define NFGRP   (N_ATTRS / FPT)        // 16 feature groups
#define QPART   8                      // i-partitions
#define BLOCK   (NFGRP * QPART)        // 128 threads
#define IPQ     (N_NODES / QPART)      // 256 i's per thread

// maintain t[0] >= t[1] >= ... >= t[7]; reject with one compare, rare bubble-insert
__device__ __forceinline__ void push8(float (&t)[TOPK], float v) {
  if (v > t[TOPK - 1]) {
    t[TOPK - 1] = v;
#pragma unroll
    for (int r = TOPK - 1; r > 0; --r) {
      float hi = fmaxf(t[r - 1], t[r]);
      float lo = fminf(t[r - 1], t[r]);
      t[r - 1] = hi;
      t[r] = lo;
    }
  }
}

// merge 4 sorted descending lists of 8 (contiguous 32 floats) -> top-8 descending
__device__ __forceinline__ void merge4(const float* __restrict__ L, float (&res)[TOPK]) {
  int p0 = 0, p1 = 0, p2 = 0, p3 = 0;
#pragma unroll
  for (int r = 0; r < TOPK; ++r) {
    float c0 = (p0 < TOPK) ? L[0 * TOPK + p0] : -INFINITY;
    float c1 = (p1 < TOPK) ? L[1 * TOPK + p1] : -INFINITY;
    float c2 = (p2 < TOPK) ? L[2 * TOPK + p2] : -INFINITY;
    float c3 = (p3 < TOPK) ? L[3 * TOPK + p3] : -INFINITY;
    float m = fmaxf(fmaxf(c0, c1), fmaxf(c2, c3));
    if      (m == c0) ++p0;
    else if (m == c1) ++p1;
    else if (m == c2) ++p2;
    else              ++p3;
    res[r] = m;
  }
}

__launch_bounds__(BLOCK)
__global__ void topk_feat_kernel(const float* __restrict__ x,
                                 const float* __restrict__ adj,
                                 float* __restrict__ out) {
  __shared__ float sh_a[N_NODES];                        // adj[:, j]            8 KB
  __shared__ float sh_top[N_ATTRS * QPART * TOPK];       // 8 partials / feature 16 KB
  __shared__ float sh_m[N_ATTRS * 2 * TOPK];             // stage-1 merges        4 KB

  const int j    = blockIdx.x;
  const int tid  = threadIdx.x;
  const int fgrp = tid & (NFGRP - 1);
  const int q    = tid >> 4;
  const int f0   = fgrp * FPT;

  // ---- stage adj column j into LDS with gfx1250 async loads (ASYNCcnt) ----
  for (int s = tid; s < N_NODES; s += BLOCK) {
    unsigned lds_off = (unsigned)(size_t)(&sh_a[s]);     // low 32b = LDS offset
    const float* gp  = adj + (size_t)s * N_NODES + j;
    asm volatile("global_load_async_to_lds_b32 %0, %1, off"
                 :: "v"(lds_off), "v"(gp) : "memory");
  }
  asm volatile("s_wait_asynccnt 0" ::: "memory");
  __syncthreads();

  // ---- per-thread top-8 for 4 features over this thread's i-partition ----
  float t[FPT][TOPK];
#pragma unroll
  for (int c = 0; c < FPT; ++c)
#pragma unroll
    for (int r = 0; r < TOPK; ++r) t[c][r] = -INFINITY;

  const int iBeg = q * IPQ;
  const float* xp = x + (size_t)iBeg * N_ATTRS + f0;

  for (int ii = 0; ii < IPQ; ii += 4) {
    // 2-address broadcast LDS read per wave: ds_load_b128, conflict-free
    float4 a4 = *reinterpret_cast<const float4*>(&sh_a[iBeg + ii]);
    // 4 coalesced global b128 loads cover 16 (i,f) elements
    float4 x0 = *reinterpret_cast<const float4*>(xp + 0 * N_ATTRS);
    float4 x1 = *reinterpret_cast<const float4*>(xp + 1 * N_ATTRS);
    float4 x2 = *reinterpret_cast<const float4*>(xp + 2 * N_ATTRS);
    float4 x3 = *reinterpret_cast<const float4*>(xp + 3 * N_ATTRS);
    xp += 4 * N_ATTRS;

    push8(t[0], a4.x * x0.x); push8(t[1], a4.x * x0.y);
    push8(t[2], a4.x * x0.z); push8(t[3], a4.x * x0.w);

    push8(t[0], a4.y * x1.x); push8(t[1], a4.y * x1.y);
    push8(t[2], a4.y * x1.z); push8(t[3], a4.y * x1.w);

    push8(t[0], a4.z * x2.x); push8(t[1], a4.z * x2.y);
    push8(t[2], a4.z * x2.z); push8(t[3], a4.z * x2.w);

    push8(t[0], a4.w * x3.x); push8(t[1], a4.w * x3.y);
    push8(t[2], a4.w * x3.z); push8(t[3], a4.w * x3.w);
  }

  // ---- publish partial sorted lists: sh_top[f][q][r] ----
#pragma unroll
  for (int c = 0; c < FPT; ++c) {
    float* dst = &sh_top[((f0 + c) * QPART + q) * TOPK];
#pragma unroll
    for (int r = 0; r < TOPK; ++r) dst[r] = t[c][r];
  }
  __syncthreads();

  // ---- stage 1: 128 threads merge 4 lists each (8 -> 2 per feature) ----
  {
    const int f = tid & (N_ATTRS - 1);
    const int g = tid >> 6;                              // 0..1
    float res[TOPK];
    merge4(&sh_top[(f * QPART + g * 4) * TOPK], res);
    float* dst = &sh_m[(f * 2 + g) * TOPK];
#pragma unroll
    for (int r = 0; r < TOPK; ++r) dst[r] = res[r];
  }
  __syncthreads();

  // ---- stage 2: 64 threads do final 2-way merge, write out[j, :, f] ----
  if (tid < N_ATTRS) {
    const float* L = &sh_m[tid * 2 * TOPK];
    int p0 = 0, p1 = 0;
    float res[TOPK];
#pragma unroll
    for (int r = 0; r < TOPK; ++r) {
      float c0 = (p0 < TOPK) ? L[p0]        : -INFINITY;
      float c1 = (p1 < TOPK) ? L[TOPK + p1] : -INFINITY;
      float m = fmaxf(c0, c1);
      if (m == c0) ++p0; else ++p1;
      res[r] = m;
    }
    float* orow = out + (size_t)j * (TOPK + 1) * N_ATTRS;
    orow[tid] = x[(size_t)j * N_ATTRS + tid];            // out[j, 0, f] = x[j, f]
#pragma unroll
    for (int r = 0; r < TOPK; ++r)
      orow[(r + 1) * N_ATTRS + tid] = res[r];            // out[j, 1+r, f], sorted desc
  }
}

extern "C" void kernel_launch(void* const* d_in, const int* in_sizes, int n_in,
                              void* d_out, int out_size, void* d_ws, size_t ws_size,
                              hipStream_t stream) {
  const float* x   = (const float*)d_in[0];   // (2048, 64)
  const float* adj = (const float*)d_in[1];   // (2048, 2048)
  float*       out = (float*)d_out;           // (2048, 9, 64)
  (void)in_sizes; (void)n_in; (void)out_size; (void)d_ws; (void)ws_size;
  topk_feat_kernel<<<N_NODES, BLOCK, 0, stream>>>(x, adj, out);
}